// ScaledDotProductAttention_73942156968306
// MI455X (gfx1250) — compile-verified
//
#include <hip/hip_runtime.h>
#include <hip/hip_bf16.h>
#include <stdint.h>

typedef float v2f __attribute__((ext_vector_type(2)));
typedef float v8f __attribute__((ext_vector_type(8)));

#define QS_STRIDE 772   // 32 Q rows: 4-bank skew per row, 16B aligned
#define KS_STRIDE 36    // 128 K rows x 32 dims per buffer, 16B aligned rows

__device__ __forceinline__ void wait_async_le4() {
#if __has_builtin(__builtin_amdgcn_s_wait_asynccnt)
  __builtin_amdgcn_s_wait_asynccnt(4);
#else
  asm volatile("s_wait_asynccnt 0x4" ::: "memory");
#endif
}
__device__ __forceinline__ void wait_async_le0() {
#if __has_builtin(__builtin_amdgcn_s_wait_asynccnt)
  __builtin_amdgcn_s_wait_asynccnt(0);
#else
  asm volatile("s_wait_asynccnt 0x0" ::: "memory");
#endif
}

// ---------------------------------------------------------------------------
// Kernel A: S[b,q,k] = mask ? -1e9 : (Q[b,q,:] . K[b,k,:]) * scale
// grid = (32 k-chunks of 128, 32 batches), block = 256 threads (8 waves)
// K tiles stream via double-buffered GLOBAL_LOAD_ASYNC_TO_LDS_B128 (ASYNCcnt)
// ---------------------------------------------------------------------------
__global__ __launch_bounds__(256) void qk_scores_kernel(
    const float* __restrict__ Q, const float* __restrict__ K,
    const unsigned char* __restrict__ mask, float* __restrict__ S) {
  __shared__ __align__(16) float Qs[32 * QS_STRIDE];       // 98,816 B
  __shared__ __align__(16) float Ks[2][128 * KS_STRIDE];   // 2 x 18,432 B

  const int b      = blockIdx.y;
  const int kchunk = blockIdx.x;
  const int t      = threadIdx.x;
  const int lane   = t & 31;
  const int w      = t >> 5;      // wave id 0..7
  const int l16    = lane & 15;
  const int hi     = lane >> 4;   // K-slice half per WMMA fp32 layout

  // --- stage Q[b] (18x768) into LDS, zero-pad rows 18..31 ---
  {
    const float4* Qg = reinterpret_cast<const float4*>(Q + (size_t)b * 18 * 768);
    for (int i = t; i < 18 * 192; i += 256) {
      int row = i / 192, c4 = i % 192;
      *reinterpret_cast<float4*>(&Qs[row * QS_STRIDE + c4 * 4]) = Qg[i];
    }
    for (int i = t; i < 14 * 192; i += 256) {
      int row = 18 + i / 192, c4 = i % 192;
      *reinterpret_cast<float4*>(&Qs[row * QS_STRIDE + c4 * 4]) =
          make_float4(0.f, 0.f, 0.f, 0.f);
    }
  }

  const int krow0 = kchunk * 128;
  const int col0  = w * 16 + l16;        // this wave's K row (= N col) in tile
  const int c4    = t & 7;               // float4 column within 32-dim slab
  const int rb    = t >> 3;              // 0..31

  // Async issue of one 128x32 K tile into LDS buffer `buf` (4 b128 per lane)
  auto issue_tile = [&](int buf, int dc) {
    const uint32_t lds0 = (uint32_t)(size_t)(&Ks[buf][0]);  // LDS offset = low 32b
    #pragma unroll
    for (int p = 0; p < 4; ++p) {
      const int row = p * 32 + rb;
      const uint64_t ga = (uint64_t)(size_t)(
          K + (size_t)(b * 4096 + krow0 + row) * 768 + dc * 32 + c4 * 4);
      const uint32_t la = lds0 + (uint32_t)((row * KS_STRIDE + c4 * 4) * 4);
      asm volatile("global_load_async_to_lds_b128 %0, %1, off"
                   :: "v"(la), "v"(ga) : "memory");
    }
  };

  v8f acc0 = {}, acc1 = {};
  issue_tile(0, 0);

  for (int dc = 0; dc < 24; ++dc) {      // 24 slabs of 32 dims = 768
    const int cur = dc & 1;
    if (dc + 1 < 24) {
      issue_tile(1 - cur, dc + 1);       // prefetch next slab (other buffer)
      wait_async_le4();                  // oldest 4 (current buffer) done
    } else {
      wait_async_le0();
    }
    __syncthreads();                     // everyone's buffer `cur` is ready

    const float* Kb = &Ks[cur][0];
    #pragma unroll
    for (int s = 0; s < 8; ++s) {        // 8 K-steps of depth 4
      const int dl = s * 4 + 2 * hi;
      const int dq = dc * 32 + dl;
      v2f a0, a1, b0;
      a0[0] = Qs[l16 * QS_STRIDE + dq];
      a0[1] = Qs[l16 * QS_STRIDE + dq + 1];
      a1[0] = Qs[(16 + l16) * QS_STRIDE + dq];
      a1[1] = Qs[(16 + l16) * QS_STRIDE + dq + 1];
      b0[0] = Kb[col0 * KS_STRIDE + dl];
      b0[1] = Kb[col0 * KS_STRIDE + dl + 1];
      acc0 = __builtin_amdgcn_wmma_f32_16x16x4_f32(false, a0, false, b0,
                                                   (short)0, acc0, false, false);
      acc1 = __builtin_amdgcn_wmma_f32_16x16x4_f32(false, a1, false, b0,
                                                   (short)0, acc1, false, false);
    }
    __syncthreads();                     // done reading `cur` before overwrite
  }

  // --- epilogue: scale + mask, scatter C fragments to S workspace ---
  const float scale = 0.03608439182435161f;  // 1/sqrt(768)
  const int cg0 = krow0 + col0;

  #pragma unroll
  for (int r = 0; r < 8; ++r) {
    const int M0 = r + 8 * hi;        // rows of M-tile 0 (0..15)
    const int M1 = 16 + r + 8 * hi;   // rows of M-tile 1 (only 16,17 valid)
    if (M0 < 18) {
      size_t i0 = (size_t)(b * 18 + M0) * 4096 + cg0;
      S[i0] = mask[i0] ? -1e9f : acc0[r] * scale;
    }
    if (M1 < 18) {
      size_t i1 = (size_t)(b * 18 + M1) * 4096 + cg0;
      S[i1] = mask[i1] ? -1e9f : acc1[r] * scale;
    }
  }
}

// ---------------------------------------------------------------------------
// Kernel B: per-row softmax stats (max, sum-of-exp) over LK=4096
// grid = 576 rows, block = 256
// ---------------------------------------------------------------------------
__global__ __launch_bounds__(256) void softmax_stats_kernel(
    const float* __restrict__ S, float* __restrict__ stats) {
  __shared__ float red[256];
  const int row = blockIdx.x;   // b*18 + q
  const int t   = threadIdx.x;
  const float* Sr = S + (size_t)row * 4096;

  float vals[16];
  float vmax = -3.4e38f;
  #pragma unroll
  for (int i = 0; i < 16; ++i) {
    vals[i] = Sr[t + 256 * i];
    vmax = fmaxf(vmax, vals[i]);
  }
  red[t] = vmax;
  __syncthreads();
  for (int off = 128; off > 0; off >>= 1) {
    if (t < off) red[t] = fmaxf(red[t], red[t + off]);
    __syncthreads();
  }
  const float m = red[0];
  __syncthreads();

  float sum = 0.f;
  #pragma unroll
  for (int i = 0; i < 16; ++i) sum += __expf(vals[i] - m);
  red[t] = sum;
  __syncthreads();
  for (int off = 128; off > 0; off >>= 1) {
    if (t < off) red[t] += red[t + off];
    __syncthreads();
  }
  if (t == 0) {
    stats[row]       = m;       // row max
    stats[576 + row] = red[0];  // row sum of exp
  }
}

// ---------------------------------------------------------------------------
// Kernel C: attn_fc[b,k] = sum_q (W[q]/l_q) * exp(S[b,q,k]-m_q)
//           context[b,:] += sum_k attn_fc[b,k] * V[b,k,:]
// grid = (16 k-chunks of 256, 32 batches), block = 256
// ---------------------------------------------------------------------------
__global__ __launch_bounds__(256) void attn_fc_context_kernel(
    const float* __restrict__ S, const float* __restrict__ stats,
    const float* __restrict__ W, const float* __restrict__ V,
    float* __restrict__ ctx, float* __restrict__ afc) {
  __shared__ float m_s[18];
  __shared__ float wol[18];
  __shared__ float p_s[256];

  const int b  = blockIdx.y;
  const int kc = blockIdx.x;
  const int t  = threadIdx.x;

  if (t < 18) {
    m_s[t] = stats[b * 18 + t];
    wol[t] = W[t] / stats[576 + b * 18 + t];
  }
  __syncthreads();

  const int k = kc * 256 + t;
  float p = 0.f;
  #pragma unroll
  for (int q = 0; q < 18; ++q)
    p += wol[q] * __expf(S[(size_t)(b * 18 + q) * 4096 + k] - m_s[q]);

  afc[b * 4096 + k] = p;
  p_s[t] = p;
  __syncthreads();

  float a0 = 0.f, a1 = 0.f, a2 = 0.f;
  const float* Vb = V + (size_t)(b * 4096 + kc * 256) * 768;
  #pragma unroll 4
  for (int kk = 0; kk < 256; ++kk) {
    const float pk = p_s[kk];
    const float* vr = Vb + (size_t)kk * 768;
    a0 += pk * vr[t];
    a1 += pk * vr[t + 256];
    a2 += pk * vr[t + 512];
  }
  atomicAdd(&ctx[b * 768 + t],       a0);
  atomicAdd(&ctx[b * 768 + t + 256], a1);
  atomicAdd(&ctx[b * 768 + t + 512], a2);
}

// ---------------------------------------------------------------------------
extern "C" void kernel_launch(void* const* d_in, const int* in_sizes, int n_in,
                              void* d_out, int out_size, void* d_ws, size_t ws_size,
                              hipStream_t stream) {
  const float* Q            = (const float*)d_in[0];
  const float* K            = (const float*)d_in[1];
  const float* V            = (const float*)d_in[2];
  const unsigned char* mask = (const unsigned char*)d_in[3];  // bool -> 1 byte
  const float* W            = (const float*)d_in[4];

  float* out = (float*)d_out;
  float* ctx = out;              // context: 32*768 floats
  float* afc = out + 32 * 768;   // attn_fc: 32*4096 floats

  float* S     = (float*)d_ws;                // 32*18*4096 floats (9.4 MB, L2-resident)
  float* stats = S + (size_t)32 * 18 * 4096;  // 2*576 floats

  qk_scores_kernel<<<dim3(32, 32), 256, 0, stream>>>(Q, K, mask, S);
  softmax_stats_kernel<<<576, 256, 0, stream>>>(S, stats);
  (void)hipMemsetAsync(ctx, 0, 32 * 768 * sizeof(float), stream);
  attn_fc_context_kernel<<<dim3(16, 32), 256, 0, stream>>>(S, stats, W, V, ctx, afc);
}